// STGAT_18021682774844
// MI455X (gfx1250) — compile-verified
//
#include <hip/hip_runtime.h>
#include <math.h>

// ---------------- problem constants ----------------
#define BATCH    64
#define N_NODES  1000
#define SEQ      12
#define FEAT     4
#define IN_CH    48
#define HEADS    2
#define H1       256
#define H2       128
#define NTOT     (BATCH * N_NODES)      // 64000
#define NEDGE    1024000L

typedef __attribute__((ext_vector_type(2))) float v2f;
typedef __attribute__((ext_vector_type(8))) float v8f;

// ---------------- generic fp32 WMMA GEMM, register-blocked ----------------
// One wave computes a TM x TN grid of 16x16 tiles (V_WMMA_F32_16X16X4_F32).
// C[M x N] = A[M x K] * B[K x N]  (+ bias[n]) (+ Cin tile)
// A frag (16x4): lane L: M=L&15, VGPR0=A[M][k+2*hi], VGPR1=A[M][k+2*hi+1]
// B frag (4x16): lane L: N=L&15, VGPR0=B[k+2*hi][N], VGPR1=B[k+2*hi+1][N]
// C/D   (16x16): lane L: N=L&15, VGPR r = C[r + 8*hi][N]
template <int TM, int TN, bool BIAS, bool CIN>
__global__ void gemm_f32_wmma(const float* __restrict__ A, const float* __restrict__ B,
                              const float* __restrict__ bias, const float* __restrict__ Cin,
                              float* __restrict__ C,
                              int M, int N, int K, int lda, int ldb, int ldcin, int ldc)
{
    const int lane   = threadIdx.x & 31;
    const int wave   = (int)((blockIdx.x * blockDim.x + threadIdx.x) >> 5);
    const int superN = N / (16 * TN);
    const int superM = M / (16 * TM);
    if (wave >= superM * superN) return;          // wave-uniform exit: EXEC stays all-ones
    const int sm = wave / superN;
    const int sn = wave - sm * superN;
    const int lo = lane & 15;
    const int hi = lane >> 4;

    v8f c[TM][TN];
#pragma unroll
    for (int i = 0; i < TM; ++i)
#pragma unroll
        for (int j = 0; j < TN; ++j)
            c[i][j] = (v8f){};

    if (CIN) {
#pragma unroll
        for (int i = 0; i < TM; ++i)
#pragma unroll
            for (int j = 0; j < TN; ++j)
#pragma unroll
                for (int r = 0; r < 8; ++r)
                    c[i][j][r] = Cin[(long)((sm * TM + i) * 16 + r + 8 * hi) * ldcin
                                     + (sn * TN + j) * 16 + lo];
    }

    const float* __restrict__ Arow[TM];
#pragma unroll
    for (int i = 0; i < TM; ++i)
        Arow[i] = A + (long)((sm * TM + i) * 16 + lo) * lda + 2 * hi;
    const float* __restrict__ Bcol[TN];
#pragma unroll
    for (int j = 0; j < TN; ++j)
        Bcol[j] = B + (long)(2 * hi) * ldb + (sn * TN + j) * 16 + lo;

    for (int k = 0; k < K; k += 4) {
        if (k + 64 < K)  // speculative prefetch of the strided B stream -> global_prefetch_b8
            __builtin_prefetch(Bcol[0] + (long)(k + 64) * ldb, 0, 0);
        v2f a[TM];
#pragma unroll
        for (int i = 0; i < TM; ++i)
            a[i] = (v2f){ Arow[i][k], Arow[i][k + 1] };
        v2f b[TN];
#pragma unroll
        for (int j = 0; j < TN; ++j)
            b[j] = (v2f){ Bcol[j][(long)k * ldb], Bcol[j][(long)(k + 1) * ldb] };
#pragma unroll
        for (int i = 0; i < TM; ++i)
#pragma unroll
            for (int j = 0; j < TN; ++j)
                c[i][j] = __builtin_amdgcn_wmma_f32_16x16x4_f32(false, a[i], false, b[j],
                                                                (short)0, c[i][j],
                                                                false, false);
    }

    if (BIAS) {
#pragma unroll
        for (int j = 0; j < TN; ++j) {
            const float bv = bias[(sn * TN + j) * 16 + lo];
#pragma unroll
            for (int i = 0; i < TM; ++i)
#pragma unroll
                for (int r = 0; r < 8; ++r)
                    c[i][j][r] += bv;
        }
    }
#pragma unroll
    for (int i = 0; i < TM; ++i)
#pragma unroll
        for (int j = 0; j < TN; ++j)
#pragma unroll
            for (int r = 0; r < 8; ++r)
                C[(long)((sm * TM + i) * 16 + r + 8 * hi) * ldc
                  + (sn * TN + j) * 16 + lo] = c[i][j][r];
}

// ---------------- edge-index dtype detection ----------------
// int64 little-endian with values < 2^31  =>  every odd 32-bit word is zero.
__global__ void detect_idx(const unsigned int* __restrict__ ei, int* __restrict__ flag)
{
    __shared__ int anynz;
    if (threadIdx.x == 0) anynz = 0;
    __syncthreads();
    if (ei[2 * threadIdx.x + 1] != 0u) atomicOr(&anynz, 1);
    __syncthreads();
    if (threadIdx.x == 0) *flag = (anynz == 0) ? 1 : 0;   // 1 = int64, 0 = int32
}

__device__ __forceinline__ long edge_at(const int* e32, const long long* e64,
                                        int f, long i)
{
    return f ? (long)e64[i] : (long)e32[i];
}

// ---------------- GAT per-node terms + self-loop init ----------------
__global__ void gat_node_terms(const float* __restrict__ H,
                               const float* __restrict__ attS,
                               const float* __restrict__ attD,
                               float* __restrict__ asrc, float* __restrict__ adst,
                               float* __restrict__ denom, float* __restrict__ acc)
{
    int tid = blockIdx.x * blockDim.x + threadIdx.x;
    if (tid >= NTOT * HEADS) return;
    const int n = tid >> 1, h = tid & 1;
    const float* __restrict__ Hp = H + (long)n * (HEADS * IN_CH) + h * IN_CH;
    const float* __restrict__ sp = attS + h * IN_CH;
    const float* __restrict__ dp = attD + h * IN_CH;
    float as = 0.f, ad = 0.f;
#pragma unroll 4
    for (int c = 0; c < IN_CH; ++c) { const float v = Hp[c]; as += v * sp[c]; ad += v * dp[c]; }
    asrc[tid] = as;
    adst[tid] = ad;
    float e = as + ad;
    e = e > 0.f ? e : 0.2f * e;
    const float w = expf(e);
    denom[tid] = w;
    float* __restrict__ ap = acc + (long)n * (HEADS * IN_CH) + h * IN_CH;
#pragma unroll 4
    for (int c = 0; c < IN_CH; ++c) ap[c] = Hp[c] * w;
}

// ---------------- edge pass 1: weights + denominator ----------------
__global__ void gat_edge_w(const int* __restrict__ e32, const long long* __restrict__ e64,
                           const int* __restrict__ flag,
                           const float* __restrict__ asrc, const float* __restrict__ adst,
                           float* __restrict__ wbuf, float* __restrict__ denom)
{
    long e = (long)blockIdx.x * blockDim.x + threadIdx.x;
    if (e >= NEDGE) return;
    const int f = *flag;
    const long s = edge_at(e32, e64, f, e);
    const long d = edge_at(e32, e64, f, NEDGE + e);
#pragma unroll
    for (int h = 0; h < HEADS; ++h) {
        float ee = asrc[s * 2 + h] + adst[d * 2 + h];
        ee = ee > 0.f ? ee : 0.2f * ee;
        const float w = expf(ee);
        wbuf[e * 2 + h] = w;
        atomicAdd(&denom[d * 2 + h], w);
    }
}

// ---------------- edge pass 2: message scatter (un-normalized) ----------------
__global__ void gat_edge_msg(const int* __restrict__ e32, const long long* __restrict__ e64,
                             const int* __restrict__ flag,
                             const float* __restrict__ Hm, const float* __restrict__ wbuf,
                             float* __restrict__ acc)
{
    long tid = (long)blockIdx.x * blockDim.x + threadIdx.x;
    if (tid >= NEDGE * 96) return;
    const long e  = tid / 96;
    const int  ch = (int)(tid - e * 96);
    const int  h  = ch >= IN_CH ? 1 : 0;
    const int f = *flag;
    const long s = edge_at(e32, e64, f, e);
    const long d = edge_at(e32, e64, f, NEDGE + e);
    atomicAdd(&acc[d * 96 + ch], Hm[s * 96 + ch] * wbuf[e * 2 + h]);
}

// ---------------- normalize, mean heads, add bias, reorder to [SEQ,B,4000] ----------------
__global__ void build_gseq(const float* __restrict__ acc, const float* __restrict__ denom,
                           const float* __restrict__ gbias, float* __restrict__ Gseq)
{
    long tid = (long)blockIdx.x * blockDim.x + threadIdx.x;
    if (tid >= (long)SEQ * BATCH * N_NODES * FEAT) return;
    const int  t = (int)(tid / (BATCH * N_NODES * FEAT));
    long r = tid - (long)t * (BATCH * N_NODES * FEAT);
    const int b = (int)(r / (N_NODES * FEAT));
    const int q = (int)(r - (long)b * (N_NODES * FEAT));
    const int node = q >> 2, fch = q & 3;
    const long n = (long)b * N_NODES + node;
    const int  c = t * FEAT + fch;
    const float d0 = denom[n * 2 + 0] + 1e-16f;
    const float d1 = denom[n * 2 + 1] + 1e-16f;
    Gseq[tid] = 0.5f * (acc[n * 96 + c] / d0 + acc[n * 96 + IN_CH + c] / d1) + gbias[c];
}

// ---------------- generic transpose: src[R x C] -> dst[C x R] ----------------
__global__ void transpose_rc(const float* __restrict__ src, float* __restrict__ dst,
                             int R, int C)
{
    long tid = (long)blockIdx.x * blockDim.x + threadIdx.x;
    if (tid >= (long)R * C) return;
    const long r = tid / C, c = tid - r * C;
    dst[c * (long)R + r] = src[tid];
}

__global__ void zero_f32(float* __restrict__ p, long n)
{
    long tid = (long)blockIdx.x * blockDim.x + threadIdx.x;
    if (tid < n) p[tid] = 0.f;
}

// ---------------- LSTM pointwise cell (gates already = xw[t] + h@Whh^T) ----------------
__global__ void lstm_cell(const float* __restrict__ g, float* __restrict__ h,
                          float* __restrict__ cst, float* __restrict__ yout, int Hs)
{
    int tid = blockIdx.x * blockDim.x + threadIdx.x;
    if (tid >= BATCH * Hs) return;
    const int b = tid / Hs, j = tid - b * Hs;
    const float* __restrict__ gb = g + (long)b * 4 * Hs;
    const float gi = 1.f / (1.f + expf(-gb[j]));
    const float gf = 1.f / (1.f + expf(-gb[Hs + j]));
    const float gg = tanhf(gb[2 * Hs + j]);
    const float go = 1.f / (1.f + expf(-gb[3 * Hs + j]));
    const float cn = gf * cst[tid] + gi * gg;
    cst[tid] = cn;
    const float hn = go * tanhf(cn);
    h[tid] = hn;
    if (yout) yout[tid] = hn;
}

// ---------------- host launch ----------------
static inline int gemm_blocks(int M, int N, int TM, int TN)
{
    return ((M / (16 * TM)) * (N / (16 * TN)) + 7) / 8;   // 8 waves / 256-thread block
}

extern "C" void kernel_launch(void* const* d_in, const int* in_sizes, int n_in,
                              void* d_out, int out_size, void* d_ws, size_t ws_size,
                              hipStream_t stream)
{
    (void)in_sizes; (void)n_in; (void)out_size; (void)ws_size;

    const float*     x       = (const float*)d_in[0];
    const int*       ei32    = (const int*)d_in[1];
    const long long* ei64    = (const long long*)d_in[1];
    const unsigned*  eiu     = (const unsigned*)d_in[1];
    const float*     W_gat   = (const float*)d_in[2];
    const float*     att_src = (const float*)d_in[3];
    const float*     att_dst = (const float*)d_in[4];
    const float*     g_bias  = (const float*)d_in[5];
    const float*     W_ih1   = (const float*)d_in[6];
    const float*     W_hh1   = (const float*)d_in[7];
    const float*     b1      = (const float*)d_in[8];
    const float*     W_ih2   = (const float*)d_in[9];
    const float*     W_hh2   = (const float*)d_in[10];
    const float*     b2      = (const float*)d_in[11];
    const float*     W_out   = (const float*)d_in[12];
    const float*     b_out   = (const float*)d_in[13];
    float*           out     = (float*)d_out;

    float* ws = (float*)d_ws;
    // region0 (phase1: H; phase2: sequence buffers) -- 6,144,000 floats
    const long R0 = 0;
    float* H    = ws + R0;
    float* Gseq = ws + R0;                       // aliases H (H dead after edge_msg)
    float* xw1  = ws + R0 + 3072000;
    float* y1   = ws + R0 + 3858432;
    float* xw2  = ws + R0 + 4055040;
    float* g1t  = ws + R0 + 4448256;
    float* g2t  = ws + R0 + 4513792;
    float* h1   = ws + R0 + 4546560;
    float* c1   = ws + R0 + 4562944;
    float* h2   = ws + R0 + 4579328;
    float* c2   = ws + R0 + 4587520;
    // region1 (phase1: acc; phase2: transposed weights) -- 6,144,000 floats
    const long R1 = 6144000;
    float* acc   = ws + R1;
    float* WihT1 = ws + R1;                      // aliases acc (acc dead after build_gseq)
    float* WhhT1 = ws + R1 + 4096000;
    float* WihT2 = ws + R1 + 4358144;
    float* WhhT2 = ws + R1 + 4489216;
    float* WoutT = ws + R1 + 4554752;
    // region2: persistent small buffers
    const long R2 = 12288000;
    float* asrc  = ws + R2;
    float* adst  = ws + R2 + 128000;
    float* denom = ws + R2 + 256000;
    float* wbuf  = ws + R2 + 384000;             // 2,048,000 floats
    int*   flag  = (int*)(ws + R2 + 2432000);

    // 1) edge-index dtype probe
    detect_idx<<<1, 256, 0, stream>>>(eiu, flag);

    // 2) H = x @ W_gat   [64000x48]@[48x96]
    gemm_f32_wmma<2, 2, false, false><<<gemm_blocks(NTOT, 96, 2, 2), 256, 0, stream>>>(
        x, W_gat, nullptr, nullptr, H, NTOT, 96, IN_CH, IN_CH, 96, 0, 96);

    // 3) per-node attention terms + self-loop initialization of denom/acc
    gat_node_terms<<<(NTOT * HEADS + 255) / 256, 256, 0, stream>>>(
        H, att_src, att_dst, asrc, adst, denom, acc);

    // 4) edge softmax weights + denominators (max-pass folded out analytically)
    gat_edge_w<<<(int)((NEDGE + 255) / 256), 256, 0, stream>>>(
        ei32, ei64, flag, asrc, adst, wbuf, denom);

    // 5) un-normalized message scatter: acc[d] += H[s] * w
    gat_edge_msg<<<(int)((NEDGE * 96 + 255) / 256), 256, 0, stream>>>(
        ei32, ei64, flag, H, wbuf, acc);

    // 6) normalize + mean heads + bias + reorder -> Gseq [12,64,4000]
    build_gseq<<<(int)(((long)SEQ * BATCH * N_NODES * FEAT + 255) / 256), 256, 0, stream>>>(
        acc, denom, g_bias, Gseq);

    // 7) transpose weights into K x N layouts (after acc is dead)
    transpose_rc<<<(4 * H1 * 4000 + 255) / 256, 256, 0, stream>>>(W_ih1, WihT1, 4 * H1, 4000);
    transpose_rc<<<(4 * H1 * H1   + 255) / 256, 256, 0, stream>>>(W_hh1, WhhT1, 4 * H1, H1);
    transpose_rc<<<(4 * H2 * H1   + 255) / 256, 256, 0, stream>>>(W_ih2, WihT2, 4 * H2, H1);
    transpose_rc<<<(4 * H2 * H2   + 255) / 256, 256, 0, stream>>>(W_hh2, WhhT2, 4 * H2, H2);
    transpose_rc<<<(2000 * H2     + 255) / 256, 256, 0, stream>>>(W_out, WoutT, 2000, H2);

    // 8) zero LSTM states (h1,c1,h2,c2 are contiguous: 49152 floats)
    zero_f32<<<(49152 + 255) / 256, 256, 0, stream>>>(h1, 49152);

    // 9) xw1 = Gseq @ W_ih1^T + b1   [768x4000]@[4000x1024]
    gemm_f32_wmma<2, 2, true, false><<<gemm_blocks(SEQ * BATCH, 4 * H1, 2, 2), 256, 0, stream>>>(
        Gseq, WihT1, b1, nullptr, xw1, SEQ * BATCH, 4 * H1, 4000, 4000, 4 * H1, 0, 4 * H1);

    // 10) LSTM layer 1
    for (int t = 0; t < SEQ; ++t) {
        gemm_f32_wmma<2, 2, false, true><<<gemm_blocks(BATCH, 4 * H1, 2, 2), 256, 0, stream>>>(
            h1, WhhT1, nullptr, xw1 + (long)t * BATCH * 4 * H1, g1t,
            BATCH, 4 * H1, H1, H1, 4 * H1, 4 * H1, 4 * H1);
        lstm_cell<<<(BATCH * H1 + 255) / 256, 256, 0, stream>>>(
            g1t, h1, c1, y1 + (long)t * BATCH * H1, H1);
    }

    // 11) xw2 = y1 @ W_ih2^T + b2   [768x256]@[256x512]
    gemm_f32_wmma<2, 2, true, false><<<gemm_blocks(SEQ * BATCH, 4 * H2, 2, 2), 256, 0, stream>>>(
        y1, WihT2, b2, nullptr, xw2, SEQ * BATCH, 4 * H2, H1, H1, 4 * H2, 0, 4 * H2);

    // 12) LSTM layer 2
    for (int t = 0; t < SEQ; ++t) {
        gemm_f32_wmma<2, 2, false, true><<<gemm_blocks(BATCH, 4 * H2, 2, 2), 256, 0, stream>>>(
            h2, WhhT2, nullptr, xw2 + (long)t * BATCH * 4 * H2, g2t,
            BATCH, 4 * H2, H2, H2, 4 * H2, 4 * H2, 4 * H2);
        lstm_cell<<<(BATCH * H2 + 255) / 256, 256, 0, stream>>>(
            g2t, h2, c2, nullptr, H2);
    }

    // 13) out = h2 @ W_out^T + b_out   [64x128]@[128x2000] -> flat [64000,2]
    gemm_f32_wmma<2, 1, true, false><<<gemm_blocks(BATCH, 2000, 2, 1), 256, 0, stream>>>(
        h2, WoutT, b_out, nullptr, out, BATCH, 2000, H2, H2, 2000, 0, 2000);
}